// WaveletConv_20237885899277
// MI455X (gfx1250) — compile-verified
//
#include <hip/hip_runtime.h>

typedef float v2f __attribute__((ext_vector_type(2)));
typedef float v8f __attribute__((ext_vector_type(8)));

#define C_CH 64
#define IMG_H 256
#define IMG_W 256
#define POS_TILE 32   // w2 positions per workgroup
#define S_PAD 40      // padded position stride: 2*S_PAD mod 64 == 16 -> conflict-free B-frag halves

__launch_bounds__(256)
__global__ void wavelet_mix_kernel(
    const float* __restrict__ x,
    const float* __restrict__ wLL, const float* __restrict__ bLL,
    const float* __restrict__ wLH, const float* __restrict__ bLH,
    const float* __restrict__ wHL, const float* __restrict__ bHL,
    const float* __restrict__ wHH, const float* __restrict__ bHH,
    float* __restrict__ out)
{
    __shared__ float sub[4 * C_CH * S_PAD];

    const int tid    = threadIdx.x;
    const int lane   = tid & 31;
    const int wave   = tid >> 5;
    const int w2base = blockIdx.x * POS_TILE;   // 0,32,64,96
    const int h2     = blockIdx.y;              // 0..127
    const int batch  = blockIdx.z;              // 0..7

    const size_t img = (size_t)IMG_H * IMG_W;
    const float* xb  = x + (size_t)batch * C_CH * img;

    // ---------------- Phase 1: Haar forward -> LDS subbands ----------------
    // 64 channels x 32 positions = 2048 items, 8 per thread.
    // Per item: two coalesced float2 loads (row pair), 4 subbands to LDS.
    #pragma unroll
    for (int i = 0; i < 8; ++i) {
        const int item = tid + i * 256;
        const int c    = item >> 5;   // channel 0..63 (constant per wave)
        const int pos  = item & 31;   // == lane -> conflict-free LDS stores
        const int w2   = w2base + pos;
        const float* p0 = xb + (size_t)c * img + (size_t)(2 * h2) * IMG_W + 2 * w2;
        const v2f r0 = *(const v2f*)p0;            // a, b   (row 2h2)
        const v2f r1 = *(const v2f*)(p0 + IMG_W);  // c, d   (row 2h2+1)
        const float a = r0.x, b_ = r0.y, c_ = r1.x, d_ = r1.y;
        const float ll = (a + b_ + c_ + d_) * 0.5f;
        const float lh = (a + b_ - c_ - d_) * 0.5f;
        const float hl = (a - b_ + c_ - d_) * 0.5f;
        const float hh = (a - b_ - c_ + d_) * 0.5f;
        const int base = c * S_PAD + pos;
        sub[0 * C_CH * S_PAD + base] = ll;
        sub[1 * C_CH * S_PAD + base] = lh;
        sub[2 * C_CH * S_PAD + base] = hl;
        sub[3 * C_CH * S_PAD + base] = hh;
    }

    // ---------------- Phase 2: preload weight A-fragments -------------------
    // A-matrix 16x4 f32 layout (ISA 7.12.2): lanes 0-15 -> M, lanes 16-31 -> M,
    // VGPR0 = K=0 (lo half) / K=2 (hi half), VGPR1 = K=1 / K=3.
    const int mtile = wave & 3;           // out-channel tile 0..3
    const int ntile = wave >> 2;          // position tile 0..1
    const int lo16  = lane & 15;
    const int hi    = lane >> 4;          // 0 or 1
    const int row   = mtile * 16 + lo16;  // out channel for A-fragment

    const float* Wb[4] = { wLL, wLH, wHL, wHH };
    v2f afrag[4][16];
    #pragma unroll
    for (int band = 0; band < 4; ++band) {
        const float* wp = Wb[band] + (size_t)row * C_CH + 2 * hi;
        #pragma unroll
        for (int k = 0; k < 16; ++k) {
            afrag[band][k] = *(const v2f*)(wp + 4 * k);  // K = 4k+2hi, 4k+2hi+1
        }
    }

    __syncthreads();

    // ---------------- Phase 3: WMMA accumulation ----------------------------
    // D(16 outch x 16 pos) += A(16x4 weights) * B(4x16 subband), 16 K-steps, 4 bands.
    v8f acc0 = {}, acc1 = {}, acc2 = {}, acc3 = {};
    const int col = ntile * 16 + lo16;    // position column for B / D
    #pragma unroll
    for (int k = 0; k < 16; ++k) {
        const int ch = 4 * k + 2 * hi;    // input-channel pair for this half-wave
        const int sb = ch * S_PAD + col;
        v2f bf0, bf1, bf2, bf3;
        bf0.x = sub[0 * C_CH * S_PAD + sb];
        bf0.y = sub[0 * C_CH * S_PAD + sb + S_PAD];
        bf1.x = sub[1 * C_CH * S_PAD + sb];
        bf1.y = sub[1 * C_CH * S_PAD + sb + S_PAD];
        bf2.x = sub[2 * C_CH * S_PAD + sb];
        bf2.y = sub[2 * C_CH * S_PAD + sb + S_PAD];
        bf3.x = sub[3 * C_CH * S_PAD + sb];
        bf3.y = sub[3 * C_CH * S_PAD + sb + S_PAD];
        acc0 = __builtin_amdgcn_wmma_f32_16x16x4_f32(false, afrag[0][k], false, bf0,
                                                     (short)0, acc0, false, false);
        acc1 = __builtin_amdgcn_wmma_f32_16x16x4_f32(false, afrag[1][k], false, bf1,
                                                     (short)0, acc1, false, false);
        acc2 = __builtin_amdgcn_wmma_f32_16x16x4_f32(false, afrag[2][k], false, bf2,
                                                     (short)0, acc2, false, false);
        acc3 = __builtin_amdgcn_wmma_f32_16x16x4_f32(false, afrag[3][k], false, bf3,
                                                     (short)0, acc3, false, false);
    }

    // ---------------- Phase 4: bias + inverse Haar + store ------------------
    // C/D layout: VGPR r -> M = r (lanes 0-15) or r+8 (lanes 16-31); N = lane&15.
    const float* Bb[4] = { bLL, bLH, bHL, bHH };
    float* ob = out + (size_t)batch * C_CH * img;
    const int w2 = w2base + col;
    #pragma unroll
    for (int r = 0; r < 8; ++r) {
        const int o = mtile * 16 + hi * 8 + r;   // output channel
        const float ll = acc0[r] + Bb[0][o];
        const float lh = acc1[r] + Bb[1][o];
        const float hl = acc2[r] + Bb[2][o];
        const float hh = acc3[r] + Bb[3][o];
        const v2f top = { (ll + lh + hl + hh) * 0.5f,    // (2h2,   2w2) , (2h2,   2w2+1)
                          (ll + lh - hl - hh) * 0.5f };
        const v2f bot = { (ll - lh + hl - hh) * 0.5f,    // (2h2+1, 2w2) , (2h2+1, 2w2+1)
                          (ll - lh - hl + hh) * 0.5f };
        float* po = ob + (size_t)o * img + (size_t)(2 * h2) * IMG_W + 2 * w2;
        *(v2f*)po            = top;
        *(v2f*)(po + IMG_W)  = bot;
    }
}

extern "C" void kernel_launch(void* const* d_in, const int* in_sizes, int n_in,
                              void* d_out, int out_size, void* d_ws, size_t ws_size,
                              hipStream_t stream) {
    (void)in_sizes; (void)n_in; (void)out_size; (void)d_ws; (void)ws_size;
    const float* x   = (const float*)d_in[0];
    const float* wLL = (const float*)d_in[1];
    const float* bLL = (const float*)d_in[2];
    const float* wLH = (const float*)d_in[3];
    const float* bLH = (const float*)d_in[4];
    const float* wHL = (const float*)d_in[5];
    const float* bHL = (const float*)d_in[6];
    const float* wHH = (const float*)d_in[7];
    const float* bHH = (const float*)d_in[8];
    float* out = (float*)d_out;

    dim3 grid(128 / POS_TILE, 128, 8);  // (w2 tiles, h2, batch) = (4,128,8)
    dim3 block(256);                    // 8 wave32
    wavelet_mix_kernel<<<grid, block, 0, stream>>>(
        x, wLL, bLL, wLH, bLH, wHL, bHL, wHH, bHH, out);
}